// GCN_85650237816963
// MI455X (gfx1250) — compile-verified
//
#include <hip/hip_runtime.h>

#define N_NODES   100000
#define N_EDGES   3200000
#define N_FEAT    512
#define N_HID     256
#define N_CLASSES 40

typedef float v2f __attribute__((ext_vector_type(2)));
typedef float v8f __attribute__((ext_vector_type(8)));

static __device__ __forceinline__ v8f wmma_f32(v2f a, v2f b, v8f c) {
    return __builtin_amdgcn_wmma_f32_16x16x4_f32(false, a, false, b, (short)0, c,
                                                 false, false);
}

// ---------------------------------------------------------------------------
// Pack W1 [512x256] -> P1 [64 kchunks][16 ntiles][32 lanes][4 floats]
// so each lane's B fragments for two consecutive K-steps are one b128 load.
// float j (0..3): s = j>>1 (K-step within chunk), v = j&1 (fragment elem)
// krow = 8c + 4s + 2*half + v ; col = t*16 + m
// ---------------------------------------------------------------------------
__global__ __launch_bounds__(256) void pack_w1(const float* __restrict__ W1,
                                               float* __restrict__ P1) {
    int o = blockIdx.x * 256 + threadIdx.x;           // 131072 elements
    if (o >= N_FEAT * N_HID) return;
    int j = o & 3, l = (o >> 2) & 31, t = (o >> 7) & 15, c = o >> 11;
    int krow = 8 * c + 4 * (j >> 1) + 2 * (l >> 4) + (j & 1);
    int col  = t * 16 + (l & 15);
    P1[o] = W1[krow * N_HID + col];
}

// Pack W2 [256x40] -> P2 [32 kchunks][3 ntiles][32 lanes][4], zero-padded to 48
__global__ __launch_bounds__(256) void pack_w2(const float* __restrict__ W2,
                                               float* __restrict__ P2) {
    int o = blockIdx.x * 256 + threadIdx.x;           // 12288 elements
    if (o >= (N_HID / 8) * 3 * 32 * 4) return;
    int c  = o / 384;
    int r  = o - c * 384;
    int t  = r >> 7;
    int r2 = r & 127;
    int l  = r2 >> 2, j = r2 & 3;
    int krow = 8 * c + 4 * (j >> 1) + 2 * (l >> 4) + (j & 1);
    int col  = t * 16 + (l & 15);
    P2[o] = (col < N_CLASSES) ? W2[krow * N_CLASSES + col] : 0.0f;
}

// ---------------------------------------------------------------------------
// GEMM1: H[100000x256] = X[100000x512] @ W1
// One wave computes TWO 16-row strips x 256 cols (32 accumulator tiles).
// K chunked by 8: per chunk 16x global_load_b128 (B) + 4x b64 (A), 64 WMMA.
// ---------------------------------------------------------------------------
__global__ __launch_bounds__(128) void gcn_gemm1(const float* __restrict__ X,
                                                 const float* __restrict__ P1,
                                                 float* __restrict__ H) {
    const int lane = threadIdx.x & 31;
    const int wave = threadIdx.x >> 5;
    const int job  = blockIdx.x * 4 + wave;           // 3125 jobs (32 rows each)
    if (job * 32 >= N_NODES) return;                  // whole-wave exit only
    const int  half = lane >> 4;
    const int  m    = lane & 15;
    const long rows = (long)job * 32;

    v8f acc0[16] = {};
    v8f acc1[16] = {};

    const float*  aR0 = X + (rows + m) * N_FEAT + 2 * half;
    const float*  aR1 = aR0 + 16 * N_FEAT;
    const float4* bP  = (const float4*)P1 + lane;

    for (int c = 0; c < N_FEAT / 8; ++c) {
        const int k = c * 8;
        v2f a00, a01, a10, a11;
        a00.x = aR0[k];     a00.y = aR0[k + 1];       // tile0, K-step 0
        a01.x = aR0[k + 4]; a01.y = aR0[k + 5];       // tile0, K-step 1
        a10.x = aR1[k];     a10.y = aR1[k + 1];       // tile1, K-step 0
        a11.x = aR1[k + 4]; a11.y = aR1[k + 5];       // tile1, K-step 1
        #pragma unroll
        for (int t = 0; t < 16; ++t) {
            float4 q = bP[(c * 16 + t) * 32];
            v2f b0; b0.x = q.x; b0.y = q.y;
            v2f b1; b1.x = q.z; b1.y = q.w;
            acc0[t] = wmma_f32(a00, b0, acc0[t]);
            acc0[t] = wmma_f32(a01, b1, acc0[t]);
            acc1[t] = wmma_f32(a10, b0, acc1[t]);
            acc1[t] = wmma_f32(a11, b1, acc1[t]);
        }
    }

    #pragma unroll
    for (int t = 0; t < 16; ++t) {
        #pragma unroll
        for (int v = 0; v < 8; ++v) {
            H[(rows + v + 8 * half) * N_HID + t * 16 + m]      = acc0[t][v];
            H[(rows + 16 + v + 8 * half) * N_HID + t * 16 + m] = acc1[t][v];
        }
    }
}

// ---------------------------------------------------------------------------
// SpMM scatter: AGG[rows[e]] += vals[e] * H[cols[e]]
// One wave per edge; each lane owns 8 features (2x float4 gather from
// L2-resident H, 8x global_atomic_add_f32 scatter).
// ---------------------------------------------------------------------------
__global__ __launch_bounds__(256) void gcn_spmm(const float* __restrict__ H,
                                                const float* __restrict__ vals,
                                                const int* __restrict__ rows,
                                                const int* __restrict__ cols,
                                                float* __restrict__ AGG) {
    const int  wave = threadIdx.x >> 5;
    const int  lane = threadIdx.x & 31;
    const long e    = (long)blockIdx.x * 8 + wave;
    if (e >= N_EDGES) return;

    const int   c = cols[e];
    const int   r = rows[e];
    const float v = vals[e];

    const float4* hp = (const float4*)(H + (long)c * N_HID + lane * 8);
    float4 h0 = hp[0];
    float4 h1 = hp[1];

    float* ap = AGG + (long)r * N_HID + lane * 8;
    atomicAdd(ap + 0, v * h0.x);
    atomicAdd(ap + 1, v * h0.y);
    atomicAdd(ap + 2, v * h0.z);
    atomicAdd(ap + 3, v * h0.w);
    atomicAdd(ap + 4, v * h1.x);
    atomicAdd(ap + 5, v * h1.y);
    atomicAdd(ap + 6, v * h1.z);
    atomicAdd(ap + 7, v * h1.w);
}

// ---------------------------------------------------------------------------
// GEMM2: out[100000x40] = relu(AGG + b1) @ W2 + b2
// ReLU + b1 fused into A-fragment load; W2 pre-packed/zero-padded to 48 cols.
// ---------------------------------------------------------------------------
__global__ __launch_bounds__(128) void gcn_gemm2(const float* __restrict__ AGG,
                                                 const float* __restrict__ b1,
                                                 const float* __restrict__ P2,
                                                 const float* __restrict__ b2,
                                                 float* __restrict__ out) {
    const int lane  = threadIdx.x & 31;
    const int wave  = threadIdx.x >> 5;
    const int mtile = blockIdx.x * 4 + wave;
    if (mtile * 16 >= N_NODES) return;
    const int  half = lane >> 4;
    const int  m    = lane & 15;
    const long row  = (long)mtile * 16;

    v8f acc[3] = {};

    const float*  aRow = AGG + (row + m) * N_HID + 2 * half;
    const float*  bRow = b1 + 2 * half;
    const float4* bP   = (const float4*)P2 + lane;

    for (int c = 0; c < N_HID / 8; ++c) {
        const int k = c * 8;
        v2f a0, a1;
        a0.x = fmaxf(aRow[k]     + bRow[k],     0.0f);   // relu(agg + b1)
        a0.y = fmaxf(aRow[k + 1] + bRow[k + 1], 0.0f);
        a1.x = fmaxf(aRow[k + 4] + bRow[k + 4], 0.0f);
        a1.y = fmaxf(aRow[k + 5] + bRow[k + 5], 0.0f);
        #pragma unroll
        for (int t = 0; t < 3; ++t) {
            float4 q = bP[(c * 3 + t) * 32];
            v2f b0; b0.x = q.x; b0.y = q.y;
            v2f b1v; b1v.x = q.z; b1v.y = q.w;
            acc[t] = wmma_f32(a0, b0, acc[t]);
            acc[t] = wmma_f32(a1, b1v, acc[t]);
        }
    }

    #pragma unroll
    for (int t = 0; t < 3; ++t) {
        int n = t * 16 + m;
        if (n < N_CLASSES) {
            float bias = b2[n];
            #pragma unroll
            for (int v = 0; v < 8; ++v) {
                out[(row + v + 8 * half) * N_CLASSES + n] = acc[t][v] + bias;
            }
        }
    }
}

// ---------------------------------------------------------------------------
extern "C" void kernel_launch(void* const* d_in, const int* in_sizes, int n_in,
                              void* d_out, int out_size, void* d_ws, size_t ws_size,
                              hipStream_t stream) {
    (void)in_sizes; (void)n_in; (void)out_size; (void)ws_size;

    const float* x    = (const float*)d_in[0];
    const float* vals = (const float*)d_in[1];
    const float* W1   = (const float*)d_in[2];
    const float* b1   = (const float*)d_in[3];
    const float* W2   = (const float*)d_in[4];
    const float* b2   = (const float*)d_in[5];
    const int*   rows = (const int*)d_in[6];
    const int*   cols = (const int*)d_in[7];
    float*       out  = (float*)d_out;

    // Workspace: H [100000x256], AGG [100000x256], P1 [131072], P2 [12288]
    float* H   = (float*)d_ws;
    float* AGG = H + (size_t)N_NODES * N_HID;
    float* P1  = AGG + (size_t)N_NODES * N_HID;
    float* P2  = P1 + (size_t)N_FEAT * N_HID;

    hipMemsetAsync(AGG, 0, (size_t)N_NODES * N_HID * sizeof(float), stream);

    pack_w1<<<(N_FEAT * N_HID + 255) / 256, 256, 0, stream>>>(W1, P1);
    pack_w2<<<((N_HID / 8) * 3 * 32 * 4 + 255) / 256, 256, 0, stream>>>(W2, P2);

    const int n_jobs = N_NODES / 32;                      // 3125 (32 rows/wave)
    gcn_gemm1<<<(n_jobs + 3) / 4, 128, 0, stream>>>(x, P1, H);

    gcn_spmm<<<(N_EDGES + 7) / 8, 256, 0, stream>>>(H, vals, rows, cols, AGG);

    const int n_mtiles = N_NODES / 16;                    // 6250
    gcn_gemm2<<<(n_mtiles + 3) / 4, 128, 0, stream>>>(AGG, b1, P2, b2, out);
}